// GAT_55980603736125
// MI455X (gfx1250) — compile-verified
//
#include <hip/hip_runtime.h>

// GAT reference collapses (softmax over singleton head axis == 1.0 everywhere):
//   out1[b,i,:] = sum_k (X@W)[b,k,:]  = (sum_n X[b,n,:]) @ W   (broadcast over i)
//   out2        = A (verbatim copy)
// Bandwidth-bound: ~280 MB traffic @ 23.3 TB/s => ~12us floor. FLOPs negligible.
// The reduced (16x128)@(128x64) GEMM is done with V_WMMA_F32_16X16X4_F32.

typedef __attribute__((ext_vector_type(2))) float v2f;
typedef __attribute__((ext_vector_type(4))) float v4f;
typedef __attribute__((ext_vector_type(8))) float v8f;

#define BATCH   8
#define NNODES  2048
#define INDIM   128
#define OUTDIM  64
#define CHUNKS  16
#define ROWS_PER_CHUNK (NNODES / CHUNKS)   // 128

// ---------------------------------------------------------------------------
// Stage 1: partial column sums of X over the node dimension.
// grid (BATCH, CHUNKS), 128 threads (one per channel c) -> coalesced loads.
// partial[(b*CHUNKS + chunk)*128 + c] = sum over 128 rows of X[b, n, c]
// ---------------------------------------------------------------------------
__global__ void gat_xsum_partial(const float* __restrict__ X,
                                 float* __restrict__ partial) {
    const int b = blockIdx.x, chunk = blockIdx.y, c = threadIdx.x;
    const float* xp = X + ((size_t)b * NNODES + (size_t)chunk * ROWS_PER_CHUNK) * INDIM + c;
    float s = 0.0f;
#pragma unroll 8
    for (int r = 0; r < ROWS_PER_CHUNK; ++r)
        s += xp[(size_t)r * INDIM];
    partial[((size_t)b * CHUNKS + chunk) * INDIM + c] = s;
}

// ---------------------------------------------------------------------------
// Stage 2: reduce partials -> Xsum (16x128, rows 8..15 zero-padded), then
// colsum(16x64) = Xsum(16x128) @ W(128x64) using V_WMMA_F32_16X16X4_F32.
// One wave (32 threads). 4 N-tiles of 16, K marched in steps of 4 => 128 WMMAs.
// ---------------------------------------------------------------------------
__global__ void gat_colsum_wmma(const float* __restrict__ partial,
                                const float* __restrict__ W,
                                float* __restrict__ colsum) {
    __shared__ float Asum[16 * INDIM];          // 16 rows x 128 K, rows 8..15 = 0

    const int lane = threadIdx.x;               // 0..31, wave32
    // Reduce the 16 chunk partials for each of the 8*128 (b,c) pairs.
    for (int p = lane; p < BATCH * INDIM; p += 32) {
        const int b = p / INDIM, c = p % INDIM;
        float s = 0.0f;
#pragma unroll
        for (int k = 0; k < CHUNKS; ++k)
            s += partial[((size_t)b * CHUNKS + k) * INDIM + c];
        Asum[b * INDIM + c] = s;
    }
    // Zero-pad rows 8..15 (M padding for the 16x16 WMMA tile).
    for (int p = lane; p < BATCH * INDIM; p += 32)
        Asum[(BATCH + p / INDIM) * INDIM + (p % INDIM)] = 0.0f;
    __syncthreads();

    const int half = lane >> 4;                 // 0: lanes 0-15, 1: lanes 16-31
    const int l15  = lane & 15;                 // A: row M ; B/D: column N

    v8f acc0 = {}, acc1 = {}, acc2 = {}, acc3 = {};
    for (int k0 = 0; k0 < INDIM; k0 += 4) {
        const int ka = k0 + half * 2;           // lane-half selects K pair
        // A fragment (16x4 f32): lanes 0-15 M=lane K={k0,k0+1}; lanes 16-31 K={k0+2,k0+3}
        v2f a;
        a.x = Asum[l15 * INDIM + ka + 0];
        a.y = Asum[l15 * INDIM + ka + 1];
        // B fragments (4x16 f32), one per 16-wide N tile of W (row-major 128x64).
        const float* w0 = W + (size_t)ka * OUTDIM + l15;
        v2f b0, b1, b2, b3;
        b0.x = w0[0];  b0.y = w0[OUTDIM +  0];
        b1.x = w0[16]; b1.y = w0[OUTDIM + 16];
        b2.x = w0[32]; b2.y = w0[OUTDIM + 32];
        b3.x = w0[48]; b3.y = w0[OUTDIM + 48];
        // (neg_a, A, neg_b, B, c_mod, C, reuse_a, reuse_b)
        acc0 = __builtin_amdgcn_wmma_f32_16x16x4_f32(false, a, false, b0, (short)0, acc0, false, false);
        acc1 = __builtin_amdgcn_wmma_f32_16x16x4_f32(false, a, false, b1, (short)0, acc1, false, false);
        acc2 = __builtin_amdgcn_wmma_f32_16x16x4_f32(false, a, false, b2, (short)0, acc2, false, false);
        acc3 = __builtin_amdgcn_wmma_f32_16x16x4_f32(false, a, false, b3, (short)0, acc3, false, false);
    }

    // D layout: lanes 0-15 vgpr r = (M=r, N=l15); lanes 16-31 hold M=8..15 (zero pad).
    if (half == 0) {
#pragma unroll
        for (int r = 0; r < BATCH; ++r) {
            colsum[(size_t)r * OUTDIM + l15 +  0] = acc0[r];
            colsum[(size_t)r * OUTDIM + l15 + 16] = acc1[r];
            colsum[(size_t)r * OUTDIM + l15 + 32] = acc2[r];
            colsum[(size_t)r * OUTDIM + l15 + 48] = acc3[r];
        }
    }
}

// ---------------------------------------------------------------------------
// Stage 3a: broadcast colsum (8x64) to out1 (8x2048x64), 128-bit stores.
// ---------------------------------------------------------------------------
__global__ void gat_broadcast(const float* __restrict__ colsum,
                              float* __restrict__ out) {
    const int total4 = BATCH * NNODES * (OUTDIM / 4);   // 262144 b128 stores
    const int stride = gridDim.x * blockDim.x;
    for (int i = blockIdx.x * blockDim.x + threadIdx.x; i < total4; i += stride) {
        const int d4 = i % (OUTDIM / 4);
        const int b  = i / (NNODES * (OUTDIM / 4));
        const v4f v = ((const v4f*)colsum)[b * (OUTDIM / 4) + d4];
        ((v4f*)out)[i] = v;
    }
}

// ---------------------------------------------------------------------------
// Stage 3b: stream-copy A -> out2 (134 MB each way) with non-temporal b128.
// ---------------------------------------------------------------------------
__global__ void gat_copy_adj(const float* __restrict__ A,
                             float* __restrict__ out) {
    const size_t total4 = (size_t)BATCH * NNODES * NNODES / 4;  // 8388608 v4f's
    const size_t stride = (size_t)gridDim.x * blockDim.x;
    for (size_t i = (size_t)blockIdx.x * blockDim.x + threadIdx.x; i < total4; i += stride) {
        const v4f v = __builtin_nontemporal_load(((const v4f*)A) + i);
        __builtin_nontemporal_store(v, ((v4f*)out) + i);
    }
}

// ---------------------------------------------------------------------------
extern "C" void kernel_launch(void* const* d_in, const int* in_sizes, int n_in,
                              void* d_out, int out_size, void* d_ws, size_t ws_size,
                              hipStream_t stream) {
    const float* X = (const float*)d_in[0];   // (8, 2048, 128)
    const float* A = (const float*)d_in[1];   // (8, 2048, 2048)
    const float* W = (const float*)d_in[2];   // (128, 64)
    // d_in[3] = a : provably unused (softmax over size-1 head axis == 1.0)

    float* out1 = (float*)d_out;                                   // 8*2048*64
    float* out2 = out1 + (size_t)BATCH * NNODES * OUTDIM;          // 8*2048*2048

    float* partial = (float*)d_ws;                                 // 8*16*128 f32
    float* colsum  = partial + BATCH * CHUNKS * INDIM;             // 8*64 f32

    dim3 g1(BATCH, CHUNKS);
    gat_xsum_partial<<<g1, 128, 0, stream>>>(X, partial);
    gat_colsum_wmma<<<1, 32, 0, stream>>>(partial, W, colsum);
    gat_broadcast<<<512, 256, 0, stream>>>(colsum, out1);
    gat_copy_adj<<<4096, 256, 0, stream>>>(A, out2);
}